// BigraphModel_29695403884793
// MI455X (gfx1250) — compile-verified
//
#include <hip/hip_runtime.h>
#include <math.h>

#define NN 100000
#define DD 128
#define EE 600000
#define NTILES (NN / 16)              // 6250 exactly
#define GEMM_WAVES 8
#define GEMM_BLOCKS ((NTILES + GEMM_WAVES - 1) / GEMM_WAVES)

typedef float v2f __attribute__((ext_vector_type(2)));
typedef float v8f __attribute__((ext_vector_type(8)));

__device__ __forceinline__ float wave_reduce_sum(float v) {
#pragma unroll
  for (int off = 16; off > 0; off >>= 1) v += __shfl_xor(v, off, 32);
  return v;
}

// ---------------- per-node inverse norm: inv[n] = 1/max(||x_n||, 1e-8) ----
__global__ void k_inv_norm(const float* __restrict__ x, float* __restrict__ inv) {
  int node = blockIdx.x * (blockDim.x >> 5) + (threadIdx.x >> 5);
  if (node >= NN) return;                    // wave-uniform exit
  int lane = threadIdx.x & 31;
  float4 v = *(const float4*)(x + (size_t)node * DD + lane * 4);
  float s = v.x * v.x + v.y * v.y + v.z * v.z + v.w * v.w;
  s = wave_reduce_sum(s);
  if (lane == 0) inv[node] = 1.0f / fmaxf(sqrtf(s), 1e-8f);
}

// ---------------- per-edge attention: ea = attr * cos(x[src], x[dst]) -----
__global__ void k_edge_att(const float* __restrict__ x, const float* __restrict__ inv,
                           const int* __restrict__ src, const int* __restrict__ dst,
                           const float* __restrict__ attr, float* __restrict__ ea) {
  int e = blockIdx.x * (blockDim.x >> 5) + (threadIdx.x >> 5);
  if (e >= EE) return;                        // wave-uniform exit
  int lane = threadIdx.x & 31;
  int s = src[e], d = dst[e];
  float4 a = *(const float4*)(x + (size_t)s * DD + lane * 4);
  float4 b = *(const float4*)(x + (size_t)d * DD + lane * 4);
  float t = a.x * b.x + a.y * b.y + a.z * b.z + a.w * b.w;
  t = wave_reduce_sum(t);
  if (lane == 0) {
    float c = t * inv[s] * inv[d];
    ea[e] = (attr ? attr[e] : 1.0f) * c;
  }
}

// ---------------- h = mask ? x @ W^T : x  (WMMA f32 16x16x4) --------------
// One wave per 16-row tile, 8 col-tiles (full 128 cols). W staged in LDS as
// float2 pairs Wl[m][j] = {W[j][2m], W[j][2m+1]} with swizzle (j+m)&127.
__global__ __launch_bounds__(256) void k_gemm_mask(
    const float* __restrict__ x, const float* __restrict__ W,
    const unsigned char* __restrict__ mask, float* __restrict__ h) {
  __shared__ v2f lds[64 * 128];               // 64 KB

  const v2f* W2 = (const v2f*)W;              // W2[j*64 + m]
  for (int t = threadIdx.x; t < 64 * 128; t += 256) {
    int j = t >> 6, m = t & 63;
    lds[m * 128 + ((j + m) & 127)] = W2[t];
  }
  __syncthreads();

  int wave = threadIdx.x >> 5;
  int tile = blockIdx.x * GEMM_WAVES + wave;
  if (tile >= NTILES) return;                 // wave-uniform exit
  int lane = threadIdx.x & 31;
  int r = lane & 15;                          // M (for A) / N (for B,C)
  int hi = lane >> 4;                         // K-half select (A,B) / M-half (C)

  size_t rowbase = (size_t)tile * 16;
  const float* xrow = x + (rowbase + r) * DD;

  v8f acc[8] = {};                            // 8 col-tiles of 16x16 f32 C
  for (int ks = 0; ks < 32; ks++) {           // k0 = 4*ks
    int m = 2 * ks + hi;                      // float2 pair index = kb/2
    v2f a = *(const v2f*)(xrow + 4 * ks + 2 * hi);  // A[r][kb], A[r][kb+1]
#pragma unroll
    for (int jt = 0; jt < 8; jt++) {
      int j = jt * 16 + r;
      v2f b = lds[m * 128 + ((j + m) & 127)]; // B[kb][j] = W[j][kb] pair
      acc[jt] = __builtin_amdgcn_wmma_f32_16x16x4_f32(
          false, a, false, b, (short)0, acc[jt], false, false);
    }
  }

  // C layout: VGPR v -> row 8*hi + v, col r
#pragma unroll
  for (int v = 0; v < 8; v++) {
    size_t rr = rowbase + 8 * hi + v;
    bool mk = mask ? (mask[rr] != 0) : true;  // uniform across 16 lanes
    float* hrow = h + rr * DD;
    if (mk) {
#pragma unroll
      for (int jt = 0; jt < 8; jt++) hrow[jt * 16 + r] = acc[jt][v];
    } else {
      const float* xr = x + rr * DD;
#pragma unroll
      for (int jt = 0; jt < 8; jt++) hrow[jt * 16 + r] = xr[jt * 16 + r];
    }
  }
}

// ---------------- zero fill ----------------------------------------------
__global__ void k_zero(float* __restrict__ p, int n) {
  int i = blockIdx.x * blockDim.x + threadIdx.x;
  if (i < n) p[i] = 0.0f;
}

// ---------------- edge scatter: ssum[dst] += ea*h[src]; cnt[dst] += 1 -----
__global__ void k_scatter(const float* __restrict__ h, const float* __restrict__ ea,
                          const int* __restrict__ src, const int* __restrict__ dst,
                          float* __restrict__ ssum, float* __restrict__ cnt) {
  int e = blockIdx.x * (blockDim.x >> 5) + (threadIdx.x >> 5);
  if (e >= EE) return;                        // wave-uniform exit
  int lane = threadIdx.x & 31;
  int s = src[e], d = dst[e];
  float a = ea[e];
  float4 v = *(const float4*)(h + (size_t)s * DD + lane * 4);
  float* out = ssum + (size_t)d * DD + lane * 4;
  unsafeAtomicAdd(out + 0, a * v.x);
  unsafeAtomicAdd(out + 1, a * v.y);
  unsafeAtomicAdd(out + 2, a * v.z);
  unsafeAtomicAdd(out + 3, a * v.w);
  if (lane == 0) unsafeAtomicAdd(cnt + d, 1.0f);
}

// ---------------- finish: out = mask? sigmoid(mean + h + b) : mean0 + h ---
__global__ void k_finish(const float* __restrict__ h, const float* __restrict__ ssum,
                         const float* __restrict__ cnt, const unsigned char* __restrict__ mask,
                         const float* __restrict__ b, float* __restrict__ out) {
  int i = blockIdx.x * blockDim.x + threadIdx.x;
  if (i >= NN * DD) return;
  int row = i >> 7, col = i & 127;
  bool mk = mask ? (mask[row] != 0) : true;
  float hv = h[i];
  float o = (mk ? ssum[i] / fmaxf(cnt[row], 1.0f) : 0.0f) + hv;
  out[i] = mk ? 1.0f / (1.0f + expf(-(o + b[col]))) : o;
}

// ---------------- host orchestration -------------------------------------
static void run_layer(const float* cur, const int* src, const int* dst, const float* ea,
                      const unsigned char* mask, const float* W, const float* b,
                      float* h, float* ssum, float* cnt, float* out, hipStream_t stream) {
  k_gemm_mask<<<GEMM_BLOCKS, 256, 0, stream>>>(cur, W, mask, h);
  k_zero<<<(NN * DD + 255) / 256, 256, 0, stream>>>(ssum, NN * DD);
  k_zero<<<(NN + 255) / 256, 256, 0, stream>>>(cnt, NN);
  k_scatter<<<(EE + 7) / 8, 256, 0, stream>>>(h, ea, src, dst, ssum, cnt);
  k_finish<<<(NN * DD + 255) / 256, 256, 0, stream>>>(h, ssum, cnt, mask, b, out);
}

extern "C" void kernel_launch(void* const* d_in, const int* in_sizes, int n_in,
                              void* d_out, int out_size, void* d_ws, size_t ws_size,
                              hipStream_t stream) {
  const float* x      = (const float*)d_in[0];
  const float* eat_ii = (const float*)d_in[1];
  const float* eat_uu = (const float*)d_in[2];
  const float* W1     = (const float*)d_in[3];
  const float* b1     = (const float*)d_in[4];
  const float* W2     = (const float*)d_in[5];
  const float* b2     = (const float*)d_in[6];
  const float* Wu     = (const float*)d_in[7];
  const float* bu     = (const float*)d_in[8];
  const int* ei_ii    = (const int*)d_in[9];    // (2,E): src then dst
  const int* ei_uu    = (const int*)d_in[10];
  const unsigned char* mask = (const unsigned char*)d_in[11];

  const int *src_ii = ei_ii, *dst_ii = ei_ii + EE;
  const int *src_uu = ei_uu, *dst_uu = ei_uu + EE;

  float* ws = (float*)d_ws;
  float* A    = ws;                 // N*D scratch (h)
  float* B    = A + (size_t)NN * DD;
  float* C    = B + (size_t)NN * DD;
  float* cnt  = C + (size_t)NN * DD;  // N
  float* invn = cnt + NN;             // N
  float* ea   = invn + NN;            // E

  const int eb = (EE + 7) / 8, nb = (NN + 7) / 8;

  // ea1 = edge_attr_ii * cos(x[src], x[dst])
  k_inv_norm<<<nb, 256, 0, stream>>>(x, invn);
  k_edge_att<<<eb, 256, 0, stream>>>(x, invn, src_ii, dst_ii, eat_ii, ea);
  // x1 = cgat(x, ii, ea1, mask, W1, b1)  -> B
  run_layer(x, src_ii, dst_ii, ea, mask, W1, b1, A, B, cnt, B, stream);
  // ea2 = edge_attr_ii * cos(x1[src], x1[dst])
  k_inv_norm<<<nb, 256, 0, stream>>>(B, invn);
  k_edge_att<<<eb, 256, 0, stream>>>(B, invn, src_ii, dst_ii, eat_ii, ea);
  // x2 = cgat(x1, ii, ea2, mask, W2, b2) -> C
  run_layer(B, src_ii, dst_ii, ea, mask, W2, b2, A, C, cnt, C, stream);
  // u1 = cgat(x2, uiu, edge_attr_uiu, all, Wu, bu) -> B
  run_layer(C, src_uu, dst_uu, eat_uu, nullptr, Wu, bu, A, B, cnt, B, stream);
  // u2 = cgat(u1, uiu, edge_attr_uiu, all, Wu, bu) -> C
  run_layer(B, src_uu, dst_uu, eat_uu, nullptr, Wu, bu, A, C, cnt, C, stream);
  // out = cos(u2[src], u2[dst])
  k_inv_norm<<<nb, 256, 0, stream>>>(C, invn);
  k_edge_att<<<eb, 256, 0, stream>>>(C, invn, src_uu, dst_uu, nullptr, (float*)d_out);
}